// SSD_83889301226088
// MI455X (gfx1250) — compile-verified
//
#include <hip/hip_runtime.h>
#include <math.h>

// ---------------------------------------------------------------------------
// SSD multibox loss for gfx1250 (MI455X). Bandwidth-bound streaming problem:
//   k_match: fused IoU match + smooth-L1 + CE (single read of preg/pcls)
//   k_ohem : per-batch OHEM top-K via binary search on float bits in LDS
//   k_final: deterministic fixed-order scalar reduction
// CDNA5 paths:
//   - V_WMMA_F32_16X16X4_F32 all-ones-B trick for exact wave32 sums
//   - GLOBAL_LOAD_ASYNC_TO_LDS_B32/B128 + s_wait_asynccnt for LDS staging
//     (guarded by __has_builtin; plain-load fallback otherwise)
// No atomics anywhere -> bitwise-deterministic accumulation order.
// ---------------------------------------------------------------------------

typedef __attribute__((ext_vector_type(2))) float v2f;
typedef __attribute__((ext_vector_type(8))) float v8f;

#define IOU_T    0.5f
#define VXY      0.1f
#define VWH      0.2f
#define EPS16    9.765625e-4f
#define NEG_R    3.0f

#if __has_builtin(__builtin_amdgcn_global_load_async_to_lds_b32) && \
    __has_builtin(__builtin_amdgcn_global_load_async_to_lds_b128) && \
    __has_builtin(__builtin_amdgcn_s_wait_asynccnt)
#define HAVE_ASYNC_LDS 1
#endif

#ifdef HAVE_ASYNC_LDS
// Builtin prototypes (from hipcc diagnostics, HIP prints AS1 as __device__,
// AS3 as __shared__):
//   b32 : (AS1 int*,  AS3 int*,  imm int offset, imm int cpol)
//   b128: (AS1 int __attribute__((vector_size(16)))*, AS3 same, imm, imm)
typedef int gvec4 __attribute__((vector_size(16)));
#define AS1 __attribute__((address_space(1)))
#define AS3 __attribute__((address_space(3)))
__device__ __forceinline__ AS1 int* gptr_i32(const void* p) {
  return (AS1 int*)(uintptr_t)p;
}
__device__ __forceinline__ AS3 int* lptr_i32(void* p) {
  // generic LDS pointer: low 32 bits are the LDS offset
  return (AS3 int*)(uint32_t)(uintptr_t)p;
}
__device__ __forceinline__ AS1 gvec4* gptr_i128(const void* p) {
  return (AS1 gvec4*)(uintptr_t)p;
}
__device__ __forceinline__ AS3 gvec4* lptr_i128(void* p) {
  return (AS3 gvec4*)(uint32_t)(uintptr_t)p;
}
#endif

// WMMA f32 16x16x4, A = per-lane values (lane m -> A[m,0], lane m+16 ->
// A[m,2]; VGPR1 zero), B = all ones  =>  D[m,n] = v_m + v_{m+16}.
// Sum of D's 8 VGPRs gives half-sum per lane group; shfl_xor(16) completes it.
__device__ __forceinline__ float wave_sum32(float v) {
  v2f a; a[0] = v;    a[1] = 0.0f;
  v2f b; b[0] = 1.0f; b[1] = 1.0f;
  v8f c = {};
  c = __builtin_amdgcn_wmma_f32_16x16x4_f32(false, a, false, b,
                                            (short)0, c, false, false);
  float s = c[0] + c[1] + c[2] + c[3] + c[4] + c[5] + c[6] + c[7];
  s += __shfl_xor(s, 16, 32);
  return s;
}

// EXEC must be all-ones at the WMMA: every thread of the block calls this.
__device__ __forceinline__ float block_sum(float v, float* sred, float* sb) {
  const int tid = threadIdx.x;
  const int wid = tid >> 5;
  float w = wave_sum32(v);
  __syncthreads();
  if ((tid & 31) == 0) sred[wid] = w;
  __syncthreads();
  if (tid == 0) {
    float t = 0.0f;
    const int nw = (int)(blockDim.x >> 5);
    for (int i = 0; i < nw; ++i) t += sred[i];
    *sb = t;
  }
  __syncthreads();
  return *sb;
}

__device__ __forceinline__ float huber(float d) {
  float ad = fabsf(d);
  return (ad < 1.0f) ? 0.5f * d * d : (ad - 0.5f);
}

// ---------------------------------------------------------------------------
// One block = 256 anchors of one batch. Fused match + box + CE.
// Writes per-(batch,tile) partial sums -> fully deterministic downstream.
__global__ void __launch_bounds__(256)
k_match(const float* __restrict__ preg,   // [B,4,A]
        const float* __restrict__ pcls,   // [B,C,A]
        const float* __restrict__ anc,    // [A,4] xywh
        const float* __restrict__ gbox,   // [B,G,4] ltrb
        const int*   __restrict__ glab,   // [B,G]
        float* __restrict__ loss_neg,     // [B,A]
        float* __restrict__ part_sl1,     // [B,T]
        float* __restrict__ part_ce,      // [B,T]
        float* __restrict__ part_np,      // [B,T]
        int A, int G, int C) {
  __shared__ float sgb[64 * 4];
  __shared__ int   sgl[64];
  __shared__ float sred[8];
  __shared__ float sb[1];

  const int tid = threadIdx.x;
  const int b   = blockIdx.y;
  const int a   = blockIdx.x * blockDim.x + tid;
  const bool valid = (a < A);
  const int  ac = valid ? a : (A - 1);

#ifdef HAVE_ASYNC_LDS
  for (int i = tid; i < G * 4; i += blockDim.x)
    __builtin_amdgcn_global_load_async_to_lds_b32(
        gptr_i32(&gbox[(size_t)b * G * 4 + i]), lptr_i32(&sgb[i]), 0, 0);
  for (int i = tid; i < G; i += blockDim.x)
    __builtin_amdgcn_global_load_async_to_lds_b32(
        gptr_i32(&glab[b * G + i]), lptr_i32(&sgl[i]), 0, 0);
  __builtin_amdgcn_s_wait_asynccnt(0);
#else
  for (int i = tid; i < G * 4; i += blockDim.x)
    sgb[i] = gbox[(size_t)b * G * 4 + i];
  for (int i = tid; i < G; i += blockDim.x)
    sgl[i] = glab[b * G + i];
#endif
  __syncthreads();

  // anchor ltrb + area
  const float4 ax = reinterpret_cast<const float4*>(anc)[ac];
  const float al = ax.x - ax.z * 0.5f, at = ax.y - ax.w * 0.5f;
  const float ar = ax.x + ax.z * 0.5f, ab = ax.y + ax.w * 0.5f;
  const float areaA = (ar - al) * (ab - at);

  // match: argmax IoU over GT (first-max wins, padded GTs masked to -1)
  float best = -INFINITY; int bi = 0;
  for (int g = 0; g < G; ++g) {
    const float gl = sgb[g * 4 + 0], gt = sgb[g * 4 + 1];
    const float gr = sgb[g * 4 + 2], gB = sgb[g * 4 + 3];
    const float w  = fmaxf(fminf(gr, ar) - fmaxf(gl, al), 0.0f);
    const float h  = fmaxf(fminf(gB, ab) - fmaxf(gt, at), 0.0f);
    const float inter = w * h;
    const float uni = fmaxf((gr - gl) * (gB - gt) + areaA - inter, 1e-8f);
    float iou = inter / uni;
    iou = (sgl[g] > 0) ? iou : -1.0f;
    if (iou > best) { best = iou; bi = g; }
  }
  const bool pos = valid && (best >= IOU_T);
  const int  lab = pos ? sgl[bi] : 0;

  // box regression (SSD encode + smooth-L1)
  const float gl = sgb[bi * 4 + 0], gt = sgb[bi * 4 + 1];
  const float gr = sgb[bi * 4 + 2], gB = sgb[bi * 4 + 3];
  const float gcx = (gl + gr) * 0.5f, gcy = (gt + gB) * 0.5f;
  const float gw  = gr - gl,          gh  = gB - gt;
  const float t0 = (gcx - ax.x) / (ax.z * VXY);
  const float t1 = (gcy - ax.y) / (ax.w * VXY);
  const float t2 = logf(fmaxf(gw, 1e-6f) / ax.z) / VWH;
  const float t3 = logf(fmaxf(gh, 1e-6f) / ax.w) / VWH;
  const size_t pb = (size_t)b * 4 * A + ac;
  const float sl1 = huber(preg[pb]                 - t0) +
                    huber(preg[pb + (size_t)A]     - t1) +
                    huber(preg[pb + 2 * (size_t)A] - t2) +
                    huber(preg[pb + 3 * (size_t)A] - t3);

  // CE: online logsumexp over C classes (single streaming pass over pcls)
  const size_t cb = (size_t)b * C * A + ac;
  __builtin_prefetch(&pcls[cb], 0, 1);   // global_prefetch_b8
  float m = -INFINITY, s = 0.0f, xl = 0.0f;
  for (int c = 0; c < C; ++c) {
    const float x = pcls[cb + (size_t)c * A];
    if (c == lab) xl = x;
    if (x > m) { s = s * expf(m - x) + 1.0f; m = x; }
    else       { s += expf(x - m); }
  }
  const float ce = m + logf(s) - xl;

  if (valid) loss_neg[(size_t)b * A + a] = pos ? 0.0f : ce;

  // block reductions via WMMA (uniform control flow -> EXEC all ones)
  const float s_sl1 = block_sum(pos ? sl1 : 0.0f, sred, sb);
  const float s_ce  = block_sum(pos ? ce  : 0.0f, sred, sb);
  const float s_np  = block_sum(pos ? 1.0f : 0.0f, sred, sb);
  if (tid == 0) {
    const size_t pi = (size_t)b * gridDim.x + blockIdx.x;
    part_sl1[pi] = s_sl1;
    part_ce[pi]  = s_ce;
    part_np[pi]  = s_np;
  }
}

// ---------------------------------------------------------------------------
// One block per batch: exact top-K negative selection.
// All loss_neg >= 0, so uint bit patterns preserve float order. Binary-search
// the K-th largest threshold; topK sum = sum(x > t*) + (K - cnt_gt) * t*.
__global__ void __launch_bounds__(256)
k_ohem(const float* __restrict__ loss_neg,
       const float* __restrict__ part_sl1,
       const float* __restrict__ part_ce,
       const float* __restrict__ part_np,
       float* __restrict__ bsum_sl1,   // [B]
       float* __restrict__ bsum_np,    // [B]
       float* __restrict__ term_pos,   // [B]
       float* __restrict__ term_neg,   // [B]
       int A, int T) {
  extern __shared__ unsigned int sln[];   // A uints (~35 KB, fits WGP LDS)
  __shared__ float sred[8];
  __shared__ float sb[1];
  __shared__ float sh_np[1], sh_ce[1];
  const int tid = threadIdx.x;
  const int b   = blockIdx.x;

#ifdef HAVE_ASYNC_LDS
  if ((A & 3) == 0) {
    for (int i = tid * 4; i < A; i += (int)blockDim.x * 4)
      __builtin_amdgcn_global_load_async_to_lds_b128(
          gptr_i128(&loss_neg[(size_t)b * A + i]), lptr_i128(&sln[i]), 0, 0);
  } else {
    for (int i = tid; i < A; i += blockDim.x)
      __builtin_amdgcn_global_load_async_to_lds_b32(
          gptr_i32(&loss_neg[(size_t)b * A + i]), lptr_i32(&sln[i]), 0, 0);
  }
  __builtin_amdgcn_s_wait_asynccnt(0);
#else
  for (int i = tid; i < A; i += blockDim.x)
    sln[i] = __float_as_uint(loss_neg[(size_t)b * A + i]);
#endif

  // fixed-order per-batch partial reduction (deterministic)
  if (tid == 0) {
    float ssl = 0.0f, sce = 0.0f, snp = 0.0f;
    for (int t = 0; t < T; ++t) {
      const size_t pi = (size_t)b * T + t;
      ssl += part_sl1[pi]; sce += part_ce[pi]; snp += part_np[pi];
    }
    bsum_sl1[b] = ssl;
    bsum_np[b]  = snp;
    sh_np[0] = snp; sh_ce[0] = sce;
  }
  __syncthreads();

  const float nums_pos = fmaxf(sh_np[0], EPS16);
  int K = (int)ceilf(NEG_R * nums_pos);   // #ranks r with r < 3*nums_pos
  if (K > A) K = A;
  if (K < 1) K = 1;

  // invariant: count_ge(lo) >= K (count_ge(0) == A >= K)
  unsigned int lo = 0u, hi = 0x7f800000u;
  while (lo < hi) {                       // uniform across block
    const unsigned int mid = lo + ((hi - lo + 1u) >> 1);
    float cl = 0.0f;
    for (int i = tid; i < A; i += blockDim.x)
      cl += (sln[i] >= mid) ? 1.0f : 0.0f;
    const float cnt = block_sum(cl, sred, sb);
    if (cnt >= (float)K) lo = mid; else hi = mid - 1u;
  }
  const float tstar = __uint_as_float(lo);

  float sg = 0.0f, cg = 0.0f;
  for (int i = tid; i < A; i += blockDim.x) {
    const float x = __uint_as_float(sln[i]);
    if (x > tstar) { sg += x; cg += 1.0f; }
  }
  const float sum_gt = block_sum(sg, sred, sb);
  const float cnt_gt = block_sum(cg, sred, sb);
  if (tid == 0) {
    float neg_sum = sum_gt;
    if (tstar > 0.0f) neg_sum += ((float)K - cnt_gt) * tstar; // ties at t*
    term_pos[b] = sh_ce[0] / nums_pos;
    term_neg[b] = neg_sum  / nums_pos;
  }
}

// ---------------------------------------------------------------------------
__global__ void k_final(const float* __restrict__ bsum_sl1,
                        const float* __restrict__ bsum_np,
                        const float* __restrict__ term_pos,
                        const float* __restrict__ term_neg,
                        float* __restrict__ out, int B) {
  if (threadIdx.x == 0 && blockIdx.x == 0) {
    float tsl = 0.0f, tnp = 0.0f, sp = 0.0f, sn = 0.0f;
    for (int b = 0; b < B; ++b) {         // fixed order -> deterministic
      tsl += bsum_sl1[b]; tnp += bsum_np[b];
      sp  += term_pos[b]; sn  += term_neg[b];
    }
    const float l_box = tsl / fmaxf(tnp, 1.0f);
    out[0] = l_box + sp / (float)B + sn / (float)B;
  }
}

// ---------------------------------------------------------------------------
extern "C" void kernel_launch(void* const* d_in, const int* in_sizes, int n_in,
                              void* d_out, int out_size, void* d_ws,
                              size_t ws_size, hipStream_t stream) {
  const float* preg = (const float*)d_in[0];   // [B,4,A]
  const float* pcls = (const float*)d_in[1];   // [B,C,A]
  const float* anc  = (const float*)d_in[2];   // [A,4]
  const float* gbox = (const float*)d_in[3];   // [B,G,4]
  const int*   glab = (const int*)d_in[4];     // [B,G]

  const int A = in_sizes[2] / 4;
  const int B = in_sizes[0] / (4 * A);
  const int G = in_sizes[4] / B;
  const int C = in_sizes[1] / (B * A);
  const int T = (A + 255) / 256;

  float* loss_neg = (float*)d_ws;                   // B*A
  float* part_sl1 = loss_neg + (size_t)B * A;       // B*T
  float* part_ce  = part_sl1 + (size_t)B * T;       // B*T
  float* part_np  = part_ce  + (size_t)B * T;       // B*T
  float* bsum_sl1 = part_np  + (size_t)B * T;       // B
  float* bsum_np  = bsum_sl1 + B;                   // B
  float* term_pos = bsum_np  + B;                   // B
  float* term_neg = term_pos + B;                   // B

  dim3 g1(T, B);
  k_match<<<g1, 256, 0, stream>>>(preg, pcls, anc, gbox, glab, loss_neg,
                                  part_sl1, part_ce, part_np, A, G, C);

  const size_t shb = (size_t)A * sizeof(unsigned int);
  k_ohem<<<B, 256, shb, stream>>>(loss_neg, part_sl1, part_ce, part_np,
                                  bsum_sl1, bsum_np, term_pos, term_neg, A, T);

  k_final<<<1, 1, 0, stream>>>(bsum_sl1, bsum_np, term_pos, term_neg,
                               (float*)d_out, B);
}